// GATBlock_2714419331269
// MI455X (gfx1250) — compile-verified
//
#include <hip/hip_runtime.h>
#include <cstdint>
#include <cstddef>

#define N_NODES 50000
#define N_EDGES 800000
#define IN_CH   256
#define NHC     512      // HEADS*OUT_CH
#define OUT_CH  128
#define HEADS   4
#define NEG_SLOPE 0.2f
#define M_TILE  48       // 3 x 16-row WMMA tiles per block; 48*260*4B = 49.9KB LDS

typedef __attribute__((ext_vector_type(2))) float v2f;
typedef __attribute__((ext_vector_type(8))) float v8f;
typedef __attribute__((__vector_size__(16))) int vi4;   // b128 async payload type

// ---------------------------------------------------------------------------
// Kernel 1: xp = x @ W via V_WMMA_F32_16X16X4_F32.
// Block = 256 threads (8 waves). Block computes rows [48*bid, 48*bid+48) x all
// 512 cols. x tile staged in LDS via GLOBAL_LOAD_ASYNC_TO_LDS_B128 (ASYNCcnt
// path); each wave owns a 64-col slice and 3 row-tiles => 12 WMMA per K-step
// against 8 B loads + 3 LDS A loads.
// ---------------------------------------------------------------------------
__launch_bounds__(256)
__global__ void gat_gemm_wmma(const float* __restrict__ x,
                              const float* __restrict__ W,
                              float* __restrict__ xp) {
  __shared__ __align__(16) float xt[M_TILE][260];  // 260-pad: conflict-free A reads
  const int tid  = threadIdx.x;
  const int row0 = blockIdx.x * M_TILE;

  // Stage 48x256 x-tile (row-clamped for the tail block)
  for (int t = tid; t < M_TILE * 64; t += 256) {
    const int r  = t >> 6;
    const int c4 = t & 63;
    int gr = row0 + r; if (gr >= N_NODES) gr = N_NODES - 1;
    const float* gsrc = x + (size_t)gr * IN_CH + c4 * 4;
#if __has_builtin(__builtin_amdgcn_global_load_async_to_lds_b128)
    __builtin_amdgcn_global_load_async_to_lds_b128(
        (__attribute__((address_space(1))) vi4*)(void*)gsrc,
        (__attribute__((address_space(3))) vi4*)(void*)&xt[r][c4 * 4],
        0, 0);
#else
    const float4 v = *(const float4*)gsrc;
    xt[r][c4 * 4 + 0] = v.x; xt[r][c4 * 4 + 1] = v.y;
    xt[r][c4 * 4 + 2] = v.z; xt[r][c4 * 4 + 3] = v.w;
#endif
  }
#if __has_builtin(__builtin_amdgcn_global_load_async_to_lds_b128) && __has_builtin(__builtin_amdgcn_s_wait_asynccnt)
  __builtin_amdgcn_s_wait_asynccnt(0);
#endif
  __syncthreads();

  const int wave = tid >> 5;
  const int lane = tid & 31;
  const int m    = lane & 15;
  const int koff = (lane >> 4) << 1;       // lanes 0-15: K=0,1  lanes 16-31: K=2,3
  const int n0   = wave * 64 + m;

  v8f acc[3][4];
  #pragma unroll
  for (int rt = 0; rt < 3; ++rt)
    #pragma unroll
    for (int t = 0; t < 4; ++t) acc[rt][t] = (v8f){};

  for (int k = 0; k < IN_CH; k += 4) {
    const float* wr0 = W + (size_t)(k + koff) * NHC;
    const float* wr1 = wr0 + NHC;
    __builtin_prefetch(wr0 + 8 * NHC, 0, 1);   // stream next K chunk of W
    v2f b[4];
    #pragma unroll
    for (int t = 0; t < 4; ++t) { b[t].x = wr0[n0 + 16 * t]; b[t].y = wr1[n0 + 16 * t]; }
    #pragma unroll
    for (int rt = 0; rt < 3; ++rt) {
      v2f a = *(const v2f*)&xt[rt * 16 + m][k + koff];
      #pragma unroll
      for (int t = 0; t < 4; ++t)
        acc[rt][t] = __builtin_amdgcn_wmma_f32_16x16x4_f32(
            false, a, false, b[t], (short)0, acc[rt][t], false, false);
    }
  }

  // C/D layout: VGPR r holds M=r (lanes 0-15) and M=r+8 (lanes 16-31), N=lane&15
  const int mbase = (lane >> 4) << 3;
  #pragma unroll
  for (int rt = 0; rt < 3; ++rt) {
    #pragma unroll
    for (int r = 0; r < 8; ++r) {
      const int row = row0 + rt * 16 + mbase + r;
      if (row < N_NODES) {
        float* orow = xp + (size_t)row * NHC;
        orow[n0]      = acc[rt][0][r];
        orow[n0 + 16] = acc[rt][1][r];
        orow[n0 + 32] = acc[rt][2][r];
        orow[n0 + 48] = acc[rt][3][r];
      }
    }
  }
}

// ---------------------------------------------------------------------------
// Kernel 2: per-node attention logits a_src/a_dst[n,h] = <xp[n,h,:], att_*[h,:]>
// One wave per (n,h); lane covers 4 channels; shuffle reduction.
// ---------------------------------------------------------------------------
__launch_bounds__(256)
__global__ void gat_attn_logits(const float* __restrict__ xp,
                                const float* __restrict__ att_src,
                                const float* __restrict__ att_dst,
                                float* __restrict__ a_src,
                                float* __restrict__ a_dst) {
  const int wid  = (int)((blockIdx.x * blockDim.x + threadIdx.x) >> 5);
  const int lane = threadIdx.x & 31;
  if (wid >= N_NODES * HEADS) return;
  const int n = wid >> 2, h = wid & 3;
  const float4 v  = *(const float4*)(xp + (size_t)n * NHC + h * OUT_CH + lane * 4);
  const float4 as = *(const float4*)(att_src + h * OUT_CH + lane * 4);
  const float4 ad = *(const float4*)(att_dst + h * OUT_CH + lane * 4);
  float ps = v.x * as.x + v.y * as.y + v.z * as.z + v.w * as.w;
  float pd = v.x * ad.x + v.y * ad.y + v.z * ad.z + v.w * ad.w;
  #pragma unroll
  for (int off = 16; off; off >>= 1) {
    ps += __shfl_xor(ps, off, 32);
    pd += __shfl_xor(pd, off, 32);
  }
  if (lane == 0) { a_src[wid] = ps; a_dst[wid] = pd; }
}

// ---------------------------------------------------------------------------
// Kernel 3: init out = bias (atomic accumulation target), emax = enc(-inf),
// denom = 0.
// ---------------------------------------------------------------------------
__global__ void gat_init(float* __restrict__ out, const float* __restrict__ bias,
                         unsigned* __restrict__ emax, float* __restrict__ denom) {
  const int i = blockIdx.x * blockDim.x + threadIdx.x;
  if (i < N_NODES * OUT_CH) out[i] = bias[i & 127];
  if (i < N_NODES * HEADS)  { emax[i] = 0x007FFFFFu; denom[i] = 0.0f; }
}

// Monotonic float<->uint order encoding so segment-max lowers to
// native global_atomic_max_u32.
__device__ __forceinline__ unsigned enc_f32(float x) {
  unsigned u = __float_as_uint(x);
  return (u & 0x80000000u) ? ~u : (u | 0x80000000u);
}
__device__ __forceinline__ float dec_f32(unsigned k) {
  unsigned u = (k & 0x80000000u) ? (k ^ 0x80000000u) : ~k;
  return __uint_as_float(u);
}

__device__ __forceinline__ float edge_score(const float* a_src, const float* a_dst,
                                            int s, int d, int h) {
  float v = a_src[s * HEADS + h] + a_dst[d * HEADS + h];
  return v > 0.0f ? v : NEG_SLOPE * v;   // LeakyReLU
}

// ---------------------------------------------------------------------------
// Kernel 4: segment max over dst (encoded u32 atomic max)
// ---------------------------------------------------------------------------
__global__ void gat_edge_max(const int* __restrict__ ei,
                             const float* __restrict__ a_src,
                             const float* __restrict__ a_dst,
                             unsigned* __restrict__ emax) {
  const int t = blockIdx.x * blockDim.x + threadIdx.x;
  if (t >= N_EDGES * HEADS) return;
  const int e = t >> 2, h = t & 3;
  const int s = ei[e], d = ei[N_EDGES + e];
  atomicMax(&emax[d * HEADS + h], enc_f32(edge_score(a_src, a_dst, s, d, h)));
}

// ---------------------------------------------------------------------------
// Kernel 5: e_exp = exp(score - max[dst]); denom[dst] += e_exp
// ---------------------------------------------------------------------------
__global__ void gat_edge_exp(const int* __restrict__ ei,
                             const float* __restrict__ a_src,
                             const float* __restrict__ a_dst,
                             const unsigned* __restrict__ emax,
                             float* __restrict__ eexp,
                             float* __restrict__ denom) {
  const int t = blockIdx.x * blockDim.x + threadIdx.x;
  if (t >= N_EDGES * HEADS) return;
  const int e = t >> 2, h = t & 3;
  const int s = ei[e], d = ei[N_EDGES + e];
  const float v  = edge_score(a_src, a_dst, s, d, h);
  const float mx = dec_f32(emax[d * HEADS + h]);
  const float ee = __expf(v - mx);
  eexp[t] = ee;
  atomicAdd(&denom[d * HEADS + h], ee);
}

// ---------------------------------------------------------------------------
// Kernel 6: weighted scatter. One wave per edge, lane owns 4 channels.
// Lanes 0-3 load one head's alpha each, broadcast by shuffle. Head-sum done
// in registers (mean folded via *0.25) so each edge issues 128 f32 atomic
// adds into out (pre-seeded with bias).
// ---------------------------------------------------------------------------
__launch_bounds__(256)
__global__ void gat_scatter(const int* __restrict__ ei,
                            const float* __restrict__ xp,
                            const float* __restrict__ eexp,
                            const float* __restrict__ denom,
                            float* __restrict__ out) {
  const int wid  = (int)((blockIdx.x * blockDim.x + threadIdx.x) >> 5);
  const int lane = threadIdx.x & 31;
  if (wid >= N_EDGES) return;
  const int s = ei[wid], d = ei[N_EDGES + wid];

  float myal = 0.0f;
  if (lane < HEADS) {
    const float num = eexp[(size_t)wid * HEADS + lane];
    const float den = denom[d * HEADS + lane];
    myal = num / (den + 1e-16f) * 0.25f;
  }
  float al[HEADS];
  #pragma unroll
  for (int h = 0; h < HEADS; ++h) al[h] = __shfl(myal, h, 32);

  float4 acc = {0.f, 0.f, 0.f, 0.f};
  #pragma unroll
  for (int h = 0; h < HEADS; ++h) {
    const float4 v = *(const float4*)(xp + (size_t)s * NHC + h * OUT_CH + lane * 4);
    acc.x += al[h] * v.x; acc.y += al[h] * v.y;
    acc.z += al[h] * v.z; acc.w += al[h] * v.w;
  }
  float* o = out + (size_t)d * OUT_CH + lane * 4;
  atomicAdd(o + 0, acc.x);
  atomicAdd(o + 1, acc.y);
  atomicAdd(o + 2, acc.z);
  atomicAdd(o + 3, acc.w);
}

// ---------------------------------------------------------------------------
extern "C" void kernel_launch(void* const* d_in, const int* in_sizes, int n_in,
                              void* d_out, int out_size, void* d_ws, size_t ws_size,
                              hipStream_t stream) {
  (void)in_sizes; (void)n_in; (void)out_size; (void)ws_size;
  const float* x       = (const float*)d_in[0];
  const int*   ei      = (const int*)  d_in[1];   // [2, E]
  const float* W       = (const float*)d_in[2];
  const float* att_src = (const float*)d_in[3];
  const float* att_dst = (const float*)d_in[4];
  const float* bias    = (const float*)d_in[5];
  float* out = (float*)d_out;

  // Workspace layout
  char* ws = (char*)d_ws;
  float*    xp    = (float*)ws;                                  // N*512 f32
  float*    a_src = (float*)(ws + (size_t)N_NODES * NHC * 4);    // N*4
  float*    a_dst = a_src + (size_t)N_NODES * HEADS;             // N*4
  unsigned* emax  = (unsigned*)(a_dst + (size_t)N_NODES * HEADS);// N*4 (encoded)
  float*    denom = (float*)(emax + (size_t)N_NODES * HEADS);    // N*4
  float*    eexp  = denom + (size_t)N_NODES * HEADS;             // E*4

  gat_gemm_wmma <<<(N_NODES + M_TILE - 1) / M_TILE, 256, 0, stream>>>(x, W, xp);
  gat_attn_logits<<<(N_NODES * HEADS) / 8, 256, 0, stream>>>(xp, att_src, att_dst, a_src, a_dst);
  gat_init      <<<(N_NODES * OUT_CH + 255) / 256, 256, 0, stream>>>(out, bias, emax, denom);
  gat_edge_max  <<<(N_EDGES * HEADS + 255) / 256, 256, 0, stream>>>(ei, a_src, a_dst, emax);
  gat_edge_exp  <<<(N_EDGES * HEADS + 255) / 256, 256, 0, stream>>>(ei, a_src, a_dst, emax, eexp, denom);
  gat_scatter   <<<(N_EDGES * 32 + 255) / 256, 256, 0, stream>>>(ei, xp, eexp, denom, out);
}